// AdvancedHybridModel_23441931502233
// MI455X (gfx1250) — compile-verified
//
#include <hip/hip_runtime.h>
#include <hip/hip_bf16.h>

// ---------------------------------------------------------------------------
// Types for WMMA fragments (CDNA5 / gfx1250, wave32)
// ---------------------------------------------------------------------------
typedef __attribute__((ext_vector_type(2))) float v2f;
typedef __attribute__((ext_vector_type(8))) float v8f;

#define NQ    12
#define QDIM  4096      // 2^12
#define NL    5
#define BATCH 4096
#define BN_EPS 1e-05f
#define QSTRIDE 20      // q_out padded from 18 -> 20 so K is a multiple of 4

// ---------------------------------------------------------------------------
// GEMM: out[m][o] = act( sum_k A[m][k] * W[o][k] + bias[o] )
// A: M x Kp row-major, W: Opad x Kp row-major (Opad multiple of 16, Kp
// multiple of 4 -> the inner loop is completely branch-free: two
// global_load_b64 fragments + one v_wmma_f32_16x16x4_f32 per K-step).
// Store is bounded by Ostore (real output width). act: 0 none, 1 leaky, 2 tanh
// ---------------------------------------------------------------------------
__global__ __launch_bounds__(256) void gemm_wmma_f32_kernel(
    const float* __restrict__ A, const float* __restrict__ W,
    const float* __restrict__ bias, float* __restrict__ out,
    int M, int Kp, int Opad, int Ostore, int act)
{
  const int lane  = threadIdx.x & 31;
  const int wave  = threadIdx.x >> 5;
  const int tilesN = Opad >> 4;
  const int tilesM = M >> 4;            // M is a multiple of 16 (4096)
  const int tile   = blockIdx.x * (blockDim.x >> 5) + wave;
  if (tile >= tilesM * tilesN) return;  // wave-uniform exit: EXEC all-1s for WMMA

  const int tm   = (tile / tilesN) << 4;
  const int tn   = (tile % tilesN) << 4;
  const int half = lane >> 4;           // 0: K pair {k,k+1}, 1: {k+2,k+3}
  const int l15  = lane & 15;
  const int koff = half << 1;
  const float* __restrict__ arow = A + (size_t)(tm + l15) * Kp + koff;
  const float* __restrict__ wrow = W + (size_t)(tn + l15) * Kp + koff;

  v8f c = {};
  for (int k = 0; k < Kp; k += 4) {
    const v2f a = *reinterpret_cast<const v2f*>(arow + k);   // global_load_b64
    const v2f b = *reinterpret_cast<const v2f*>(wrow + k);   // global_load_b64
    c = __builtin_amdgcn_wmma_f32_16x16x4_f32(
        /*neg_a=*/false, a, /*neg_b=*/false, b,
        /*c_mod=*/(short)0, c, /*reuse_a=*/false, /*reuse_b=*/false);
  }

  const int n = tn + l15;               // C-frag column for this lane
  if (n < Ostore) {
    const float bv = bias[n];
    #pragma unroll
    for (int r = 0; r < 8; ++r) {
      float v = c[r] + bv;
      if (act == 1)      v = v > 0.0f ? v : 0.01f * v;
      else if (act == 2) v = tanhf(v);
      out[(size_t)(tm + r + half * 8) * Ostore + n] = v;  // C layout: vgpr r -> row r (+8 upper)
    }
  }
}

// ---------------------------------------------------------------------------
// Zero-pad a row-major Osrc x Ksrc matrix into an Opad x Kpad buffer.
// ---------------------------------------------------------------------------
__global__ __launch_bounds__(256) void pad_matrix_kernel(
    const float* __restrict__ src, float* __restrict__ dst,
    int Osrc, int Ksrc, int Opad, int Kpad)
{
  const int idx = blockIdx.x * blockDim.x + threadIdx.x;
  if (idx >= Opad * Kpad) return;
  const int o = idx / Kpad, k = idx % Kpad;
  dst[idx] = (o < Osrc && k < Ksrc) ? src[o * Ksrc + k] : 0.0f;
}

// ---------------------------------------------------------------------------
// Per-feature batch statistics: mean/var over B rows (deterministic tree)
// One block per feature.
// ---------------------------------------------------------------------------
__global__ __launch_bounds__(256) void stats_kernel(
    const float* __restrict__ y, float* __restrict__ mu, float* __restrict__ var,
    int B, int F)
{
  __shared__ float ssum[256];
  __shared__ float ssq[256];
  const int f = blockIdx.x, tid = threadIdx.x;
  float s = 0.0f, s2 = 0.0f;
  for (int b = tid; b < B; b += 256) {
    const float v = y[b * F + f];
    s += v; s2 += v * v;
  }
  ssum[tid] = s; ssq[tid] = s2;
  __syncthreads();
  for (int off = 128; off > 0; off >>= 1) {
    if (tid < off) { ssum[tid] += ssum[tid + off]; ssq[tid] += ssq[tid + off]; }
    __syncthreads();
  }
  if (tid == 0) {
    const float m = ssum[0] / (float)B;
    mu[f]  = m;
    var[f] = ssq[0] / (float)B - m * m;
  }
}

// ---------------------------------------------------------------------------
// Fused BN + leaky + optional residual:  out = leaky(g*(y-mu)*rsqrt(var+eps)+be)
//                                               + rscale * res[b*resStride + f]
// In-place safe (out may alias y).
// ---------------------------------------------------------------------------
__global__ __launch_bounds__(256) void bn_leaky_kernel(
    const float* __restrict__ y, const float* __restrict__ g,
    const float* __restrict__ be, const float* __restrict__ mu,
    const float* __restrict__ var, float* __restrict__ out,
    int B, int F, const float* __restrict__ res, int resStride, float rscale)
{
  const int idx = blockIdx.x * blockDim.x + threadIdx.x;
  if (idx >= B * F) return;
  const int b = idx / F, f = idx % F;
  float v = g[f] * (y[idx] - mu[f]) * rsqrtf(var[f] + BN_EPS) + be[f];
  v = v > 0.0f ? v : 0.01f * v;
  if (res) v += rscale * res[b * resStride + f];
  out[idx] = v;
}

// ---------------------------------------------------------------------------
// 12-qubit statevector simulator. One workgroup (256 threads) per sample.
// State: 4096 complex64 = 32 KB in LDS. 5 layers x 12 U3*RZ*RY*RX gates + CNOTs.
// Outputs 12 <Z_q> then 6 <X_q> expectations per sample (stride QSTRIDE,
// columns 18..19 zeroed so the next GEMM can use K=20).
// Qubit q maps to bit position (11-q) of the flat index (qubit 0 = MSB).
// ---------------------------------------------------------------------------
struct cplx { float re, im; };
__device__ inline cplx cmul(cplx a, cplx b) {
  return { a.re * b.re - a.im * b.im, a.re * b.im + a.im * b.re };
}
__device__ inline cplx cadd(cplx a, cplx b) { return { a.re + b.re, a.im + b.im }; }

__device__ inline void cnot_swap(float2* st, int tid, int c, int t) {
  const int pc = NQ - 1 - c, pt = NQ - 1 - t;
  const int ph = pc > pt ? pc : pt, pl = pc > pt ? pt : pc;
  const int mc = 1 << pc, mt = 1 << pt;
  for (int p = tid; p < QDIM / 4; p += 256) {
    int i = ((p >> pl) << (pl + 1)) | (p & ((1 << pl) - 1));   // insert 0 at pl
    i     = ((i >> ph) << (ph + 1)) | (i & ((1 << ph) - 1));   // insert 0 at ph
    const int i0 = i | mc;        // control=1, target=0
    const int i1 = i0 | mt;       // control=1, target=1
    const float2 a = st[i0], b = st[i1];
    st[i0] = b; st[i1] = a;
  }
}

__global__ __launch_bounds__(256) void quantum_kernel(
    const float* __restrict__ xpre, const float* __restrict__ qw,
    float* __restrict__ qout)
{
  __shared__ float2 st[QDIM];             // 32 KB statevector
  __shared__ float2 gm[NL * NQ][4];       // 60 precomputed 2x2 complex gates
  __shared__ float  cs[NQ], sn[NQ];
  __shared__ float  red[8 * 18];

  const int tid = threadIdx.x;
  const int b   = blockIdx.x;

  if (tid < NQ) {
    const float h = xpre[b * NQ + tid] * 1.5707963267948966f;  // x_pre * pi/2
    cs[tid] = cosf(h);
    sn[tid] = sinf(h);
  }
  if (tid < NL * NQ) {
    const float aa = qw[tid * 3 + 0], bb = qw[tid * 3 + 1], cc = qw[tid * 3 + 2];
    const float ca = cosf(aa * 0.5f), sa = sinf(aa * 0.5f);
    const float cb = cosf(bb * 0.5f), sb = sinf(bb * 0.5f);
    const float cz = cosf(cc * 0.5f), sz = sinf(cc * 0.5f);
    const cplx RX[4] = { {ca, 0}, {0, -sa}, {0, -sa}, {ca, 0} };
    const cplx RY[4] = { {cb, 0}, {-sb, 0}, {sb, 0},  {cb, 0} };
    const cplx z00 = { cz, -sz }, z11 = { cz, sz };             // e^{-ic/2}, e^{+ic/2}
    const cplx eib = { cosf(bb), sinf(bb) };
    const cplx eic = { cosf(cc), sinf(cc) };
    const cplx U3[4] = { {ca, 0}, cmul({ -sa, 0 }, eic),
                         cmul({ sa, 0 }, eib), cmul(cmul(eib, eic), { ca, 0 }) };
    cplx T[4], T2[4], G[4];
    T[0] = cadd(cmul(RY[0], RX[0]), cmul(RY[1], RX[2]));        // T = RY*RX
    T[1] = cadd(cmul(RY[0], RX[1]), cmul(RY[1], RX[3]));
    T[2] = cadd(cmul(RY[2], RX[0]), cmul(RY[3], RX[2]));
    T[3] = cadd(cmul(RY[2], RX[1]), cmul(RY[3], RX[3]));
    T2[0] = cmul(z00, T[0]); T2[1] = cmul(z00, T[1]);            // T2 = RZ*T
    T2[2] = cmul(z11, T[2]); T2[3] = cmul(z11, T[3]);
    G[0] = cadd(cmul(U3[0], T2[0]), cmul(U3[1], T2[2]));         // G = U3*T2
    G[1] = cadd(cmul(U3[0], T2[1]), cmul(U3[1], T2[3]));
    G[2] = cadd(cmul(U3[2], T2[0]), cmul(U3[3], T2[2]));
    G[3] = cadd(cmul(U3[2], T2[1]), cmul(U3[3], T2[3]));
    #pragma unroll
    for (int i = 0; i < 4; ++i) gm[tid][i] = make_float2(G[i].re, G[i].im);
  }
  __syncthreads();

  // Initial product state: amp(i) = prod_q (bit_q ? sin : cos)
  for (int i = tid; i < QDIM; i += 256) {
    float amp = 1.0f;
    #pragma unroll
    for (int q = 0; q < NQ; ++q)
      amp *= ((i >> (NQ - 1 - q)) & 1) ? sn[q] : cs[q];
    st[i] = make_float2(amp, 0.0f);
  }
  __syncthreads();

  for (int l = 0; l < NL; ++l) {
    for (int q = 0; q < NQ; ++q) {
      const int mmask = 1 << (NQ - 1 - q);
      const float2 u00 = gm[l * NQ + q][0], u01 = gm[l * NQ + q][1];
      const float2 u10 = gm[l * NQ + q][2], u11 = gm[l * NQ + q][3];
      for (int p = tid; p < QDIM / 2; p += 256) {
        const int i0 = ((p & ~(mmask - 1)) << 1) | (p & (mmask - 1));
        const int i1 = i0 | mmask;
        const float2 s0 = st[i0], s1 = st[i1];
        float2 r0, r1;
        r0.x = u00.x * s0.x - u00.y * s0.y + u01.x * s1.x - u01.y * s1.y;
        r0.y = u00.x * s0.y + u00.y * s0.x + u01.x * s1.y + u01.y * s1.x;
        r1.x = u10.x * s0.x - u10.y * s0.y + u11.x * s1.x - u11.y * s1.y;
        r1.y = u10.x * s0.y + u10.y * s0.x + u11.x * s1.y + u11.y * s1.x;
        st[i0] = r0; st[i1] = r1;
      }
      __syncthreads();
    }
    if ((l & 1) == 0) {
      for (int q = 0; q < NQ - 1; ++q) { cnot_swap(st, tid, q, q + 1); __syncthreads(); }
      cnot_swap(st, tid, NQ - 1, 0); __syncthreads();
    } else {
      for (int q = 0; q < NQ; q += 2) {
        cnot_swap(st, tid, q, (q + NQ / 2) % NQ); __syncthreads();
      }
    }
  }

  // Expectations: 12 x <Z_q> from probabilities, 6 x <X_q> from bit-flip pairs.
  float acc[18];
  #pragma unroll
  for (int k = 0; k < 18; ++k) acc[k] = 0.0f;
  for (int i = tid; i < QDIM; i += 256) {
    const float2 s = st[i];
    const float pr = s.x * s.x + s.y * s.y;
    #pragma unroll
    for (int q = 0; q < NQ; ++q)
      acc[q] += ((i >> (NQ - 1 - q)) & 1) ? -pr : pr;
  }
  for (int q = 0; q < 6; ++q) {
    const int mmask = 1 << (NQ - 1 - q);
    for (int p = tid; p < QDIM / 2; p += 256) {
      const int i0 = ((p & ~(mmask - 1)) << 1) | (p & (mmask - 1));
      const int i1 = i0 | mmask;
      const float2 s0 = st[i0], s1 = st[i1];
      acc[12 + q] += 2.0f * (s0.x * s1.x + s0.y * s1.y);   // 2*Re(conj(s0)*s1)
    }
  }
  // Deterministic reduce: wave32 shuffle tree, then cross-wave LDS sum.
  #pragma unroll
  for (int k = 0; k < 18; ++k) {
    float v = acc[k];
    for (int off = 16; off > 0; off >>= 1) v += __shfl_xor(v, off, 32);
    if ((tid & 31) == 0) red[(tid >> 5) * 18 + k] = v;
  }
  __syncthreads();
  if (tid < QSTRIDE) {
    float s = 0.0f;
    if (tid < 18) {
      #pragma unroll
      for (int w = 0; w < 8; ++w) s += red[w * 18 + tid];
    }
    qout[b * QSTRIDE + tid] = s;   // cols 18..19 written as 0 (K padding)
  }
}

// ---------------------------------------------------------------------------
// Host launcher
// ---------------------------------------------------------------------------
extern "C" void kernel_launch(void* const* d_in, const int* in_sizes, int n_in,
                              void* d_out, int out_size, void* d_ws, size_t ws_size,
                              hipStream_t stream) {
  (void)in_sizes; (void)n_in; (void)out_size; (void)ws_size;
  const float* x    = (const float*)d_in[0];
  const float* Ws   = (const float*)d_in[1];
  const float* bs   = (const float*)d_in[2];
  const float* W1   = (const float*)d_in[3];
  const float* b1   = (const float*)d_in[4];
  const float* g1   = (const float*)d_in[5];
  const float* be1  = (const float*)d_in[6];
  const float* W2   = (const float*)d_in[7];
  const float* b2   = (const float*)d_in[8];
  const float* g2   = (const float*)d_in[9];
  const float* be2  = (const float*)d_in[10];
  const float* W3   = (const float*)d_in[11];
  const float* b3   = (const float*)d_in[12];
  const float* g3   = (const float*)d_in[13];
  const float* be3  = (const float*)d_in[14];
  const float* Wp   = (const float*)d_in[15];
  const float* bp   = (const float*)d_in[16];
  const float* qw   = (const float*)d_in[17];
  const float* Wq1  = (const float*)d_in[18];
  const float* bq1  = (const float*)d_in[19];
  const float* gq1  = (const float*)d_in[20];
  const float* beq1 = (const float*)d_in[21];
  const float* Wq2  = (const float*)d_in[22];
  const float* bq2  = (const float*)d_in[23];
  const float* gq2  = (const float*)d_in[24];
  const float* beq2 = (const float*)d_in[25];
  const float* Wo1  = (const float*)d_in[26];
  const float* bo1  = (const float*)d_in[27];
  const float* Wo2  = (const float*)d_in[28];
  const float* bo2  = (const float*)d_in[29];
  float* out = (float*)d_out;

  const int B = BATCH;
  char* wsp = (char*)d_ws;
  auto alloc = [&](size_t n) { float* p = (float*)wsp; wsp += n * sizeof(float); return p; };
  float* skip = alloc((size_t)B * 128);
  float* y1   = alloc((size_t)B * 256);   // y1 -> x1 in place
  float* y2   = alloc((size_t)B * 128);   // y2 -> x2 in place
  float* y3   = alloc((size_t)B * 64);    // y3 -> x3 in place
  float* xpre = alloc((size_t)B * NQ);
  float* qo   = alloc((size_t)B * QSTRIDE);
  float* y4   = alloc((size_t)B * 128);   // y4 -> p1 in place
  float* y5   = alloc((size_t)B * 64);    // y5 -> p2 in place
  float* y6   = alloc((size_t)B * 32);
  float* WpP  = alloc(16 * 64);           // Wp  12x64  -> 16x64  (row pad)
  float* Wq1P = alloc(128 * QSTRIDE);     // Wq1 128x18 -> 128x20 (col pad)
  float* Wo2P = alloc(16 * 32);           // Wo2 1x32   -> 16x32  (row pad)
  float* mu1 = alloc(256); float* var1 = alloc(256);
  float* mu2 = alloc(128); float* var2 = alloc(128);
  float* mu3 = alloc(64);  float* var3 = alloc(64);
  float* mu4 = alloc(128); float* var4 = alloc(128);
  float* mu5 = alloc(64);  float* var5 = alloc(64);

  auto gemm = [&](const float* A, const float* W, const float* bias, float* o,
                  int Kp, int Opad, int Ostore, int act) {
    const int tiles  = (B / 16) * (Opad / 16);
    const int blocks = (tiles + 7) / 8;                   // 8 waves per block
    gemm_wmma_f32_kernel<<<blocks, 256, 0, stream>>>(A, W, bias, o, B, Kp,
                                                     Opad, Ostore, act);
  };
  auto pad = [&](const float* src, float* dst, int Os, int Ks, int Op, int Kp2) {
    const int n = Op * Kp2;
    pad_matrix_kernel<<<(n + 255) / 256, 256, 0, stream>>>(src, dst, Os, Ks, Op, Kp2);
  };
  auto stats = [&](const float* y, float* mu, float* var, int F) {
    stats_kernel<<<F, 256, 0, stream>>>(y, mu, var, B, F);
  };
  auto bnact = [&](const float* y, const float* g, const float* be,
                   const float* mu, const float* var, float* o, int F,
                   const float* res, int rs, float rscale) {
    const int n = B * F;
    bn_leaky_kernel<<<(n + 255) / 256, 256, 0, stream>>>(y, g, be, mu, var, o,
                                                         B, F, res, rs, rscale);
  };

  // Pad the irregular weight matrices once (tiny).
  pad(Wp,  WpP,  12, 64, 16, 64);
  pad(Wq1, Wq1P, 128, 18, 128, QSTRIDE);
  pad(Wo2, Wo2P, 1, 32, 16, 32);

  // skip = x @ Ws^T + bs
  gemm(x, Ws, bs, skip, 64, 128, 128, 0);
  // x1 = leaky(bn(x @ W1^T + b1))
  gemm(x, W1, b1, y1, 64, 256, 256, 0);
  stats(y1, mu1, var1, 256);
  bnact(y1, g1, be1, mu1, var1, y1, 256, nullptr, 0, 0.0f);
  // x2 = leaky(bn(x1 @ W2^T + b2))
  gemm(y1, W2, b2, y2, 256, 128, 128, 0);
  stats(y2, mu2, var2, 128);
  bnact(y2, g2, be2, mu2, var2, y2, 128, nullptr, 0, 0.0f);
  // x3 = leaky(bn(x2 @ W3^T + b3)) + 0.1 * x1[:, :64]
  gemm(y2, W3, b3, y3, 128, 64, 64, 0);
  stats(y3, mu3, var3, 64);
  bnact(y3, g3, be3, mu3, var3, y3, 64, y1, 256, 0.1f);
  // x_pre = tanh(x3 @ Wp^T + bp)   (Wp padded to 16 rows; store only 12 cols)
  gemm(y3, WpP, bp, xpre, 64, 16, NQ, 2);
  // q_out = quantum(x_pre, qw)  -> B x 20 (cols 18..19 zero)
  quantum_kernel<<<B, 256, 0, stream>>>(xpre, qw, qo);
  // p1 = leaky(bn(q_out @ Wq1^T + bq1)) + skip   (K padded 18 -> 20)
  gemm(qo, Wq1P, bq1, y4, QSTRIDE, 128, 128, 0);
  stats(y4, mu4, var4, 128);
  bnact(y4, gq1, beq1, mu4, var4, y4, 128, skip, 128, 1.0f);
  // p2 = leaky(bn(p1 @ Wq2^T + bq2))
  gemm(y4, Wq2, bq2, y5, 128, 64, 64, 0);
  stats(y5, mu5, var5, 64);
  bnact(y5, gq2, beq2, mu5, var5, y5, 64, nullptr, 0, 0.0f);
  // out = leaky(p2 @ Wo1^T + bo1) @ Wo2^T + bo2   (Wo2 padded to 16 rows)
  gemm(y5, Wo1, bo1, y6, 64, 32, 32, 1);
  gemm(y6, Wo2P, bo2, out, 32, 16, 1, 0);
}